// SetCriterion_TM_33328946217441
// MI455X (gfx1250) — compile-verified
//
#include <hip/hip_runtime.h>
#include <hip/hip_bf16.h>

// Problem constants from the reference.
#define Bn 16
#define Nn 32
#define Hn 512
#define Wn 512

#define ROWS 8                 // rows per block
#define THREADS 256            // 8 waves (wave32)
#define F4_PER_ROW (Wn / 4)    // 128 float4 per row
#define ITERS (ROWS * F4_PER_ROW / THREADS)   // 4

__global__ __launch_bounds__(THREADS)
void raster_loss_partial(const float* __restrict__ obj,
                         const float* __restrict__ lbrts,
                         const float* __restrict__ boxes,
                         float* __restrict__ partial) {
    __shared__ float4   rawbox[Nn];     // async-staged raw boxes (normalized x1y1x2y2)
    __shared__ unsigned xr[Nn];         // x1i | (x2i << 16)
    __shared__ unsigned yr[Nn];         // y1i | (y2i << 16), 0 if invalid
    __shared__ float4   regs[Nn];       // (l, t, r, b) per-box constant targets
    __shared__ unsigned ymask_s[ROWS];  // per-row box validity bitmask
    __shared__ float    red0[THREADS], red1[THREADS], red2[THREADS];

    const int tid  = threadIdx.x;
    const int b    = blockIdx.x / (Hn / ROWS);
    const int tile = blockIdx.x % (Hn / ROWS);
    const int y0   = tile * ROWS;

    // ---- Stage box table into LDS via CDNA5 async global->LDS copy (wave 0) ----
    if (tid < Nn) {
        unsigned lds_off = (unsigned)(size_t)(&rawbox[tid]);   // low 32 bits = LDS byte addr
        unsigned long long gaddr =
            (unsigned long long)(const void*)(boxes + ((size_t)b * Nn + tid) * 4);
        asm volatile("global_load_async_to_lds_b128 %0, %1, off"
                     :: "v"(lds_off), "v"(gaddr) : "memory");
        asm volatile("s_wait_asynccnt 0x0" ::: "memory");

        float4 bx = rawbox[tid];
        float x1 = bx.x * (float)Wn, y1 = bx.y * (float)Hn;
        float x2 = bx.z * (float)Wn, y2 = bx.w * (float)Hn;
        // int() truncation toward zero + clamps, matching the reference
        int x1i = min(max((int)x1, 0), Wn - 1);
        int y1i = min(max((int)y1, 0), Hn - 1);
        int x2i = min(max((int)x2 + 1, 0), Wn);
        int y2i = min(max((int)y2 + 1, 0), Hn);
        bool valid = (x2i > x1i) && (y2i > y1i);
        float cx = 0.5f * (x1 + x2), cy = 0.5f * (y1 + y2);
        xr[tid]   = (unsigned)x1i | ((unsigned)x2i << 16);
        yr[tid]   = valid ? ((unsigned)y1i | ((unsigned)y2i << 16)) : 0u;
        regs[tid] = make_float4(cx - (float)x1i, cy - (float)y1i,
                                (float)x2i - cx, (float)y2i - cy);
    }
    __syncthreads();

    // ---- Per-row y-validity bitmask (one 32-bit mask per stripe row) ----
    if (tid < ROWS) {
        unsigned y = (unsigned)(y0 + tid);
        unsigned m = 0;
        #pragma unroll
        for (int n = 0; n < Nn; ++n) {
            unsigned p = yr[n];
            if (y >= (p & 0xffffu) && y < (p >> 16)) m |= (1u << n);
        }
        ymask_s[tid] = m;
    }
    __syncthreads();

    // ---- Stream the 5 planes; accumulate (ce, l1, cnt) in registers ----
    float ce = 0.f, l1 = 0.f, cnt = 0.f;
    const size_t img  = (size_t)Hn * Wn;
    const float* objb = obj   + (size_t)b * img;
    const float* lb   = lbrts + (size_t)b * 4 * img;

    for (int it = 0; it < ITERS; ++it) {
        int flat = it * THREADS + tid;     // 0..1023
        int r    = flat >> 7;              // row in stripe (F4_PER_ROW == 128)
        int c    = flat & (F4_PER_ROW - 1);
        int y    = y0 + r;
        int x0p  = c << 2;
        size_t off = (size_t)y * Wn + x0p;

        // prefetch next iteration's cachelines (gfx1250 global_prefetch_b8)
        if (it + 1 < ITERS) {
            size_t noff = off + (size_t)(THREADS / F4_PER_ROW) * Wn;
            __builtin_prefetch(objb + noff,           0, 0);
            __builtin_prefetch(lb + 0 * img + noff,   0, 0);
            __builtin_prefetch(lb + 1 * img + noff,   0, 0);
            __builtin_prefetch(lb + 2 * img + noff,   0, 0);
            __builtin_prefetch(lb + 3 * img + noff,   0, 0);
        }

        float4 o4 = *(const float4*)(objb + off);
        float4 p0 = *(const float4*)(lb + 0 * img + off);
        float4 p1 = *(const float4*)(lb + 1 * img + off);
        float4 p2 = *(const float4*)(lb + 2 * img + off);
        float4 p3 = *(const float4*)(lb + 3 * img + off);

        unsigned m = ymask_s[r];
        float ov[4] = {o4.x, o4.y, o4.z, o4.w};
        float a0[4] = {p0.x, p0.y, p0.z, p0.w};
        float a1[4] = {p1.x, p1.y, p1.z, p1.w};
        float a2[4] = {p2.x, p2.y, p2.z, p2.w};
        float a3[4] = {p3.x, p3.y, p3.z, p3.w};

        #pragma unroll
        for (int j = 0; j < 4; ++j) {
            unsigned x = (unsigned)(x0p + j);
            int last = -1;
            unsigned mm = m;
            while (mm) {                       // scan MSB->LSB: last box wins
                int n = 31 - __clz((int)mm);
                unsigned p = xr[n];
                if (x >= (p & 0xffffu) && x < (p >> 16)) { last = n; break; }
                mm &= (1u << n) - 1u;          // clear current MSB, keep lower bits
            }
            float o = ov[j];
            float t = (last >= 0) ? 1.f : 0.f;
            // logaddexp(0, o) = max(o,0) + log1p(exp(-|o|))
            ce += fmaxf(o, 0.f) + log1pf(expf(-fabsf(o))) - o * t;
            if (last >= 0) {
                float4 rv = regs[last];
                l1 += fabsf(a0[j] - rv.x) + fabsf(a1[j] - rv.y) +
                      fabsf(a2[j] - rv.z) + fabsf(a3[j] - rv.w);
                cnt += 4.f;                    // mask expanded over 4 channels
            }
        }
    }

    // ---- Block tree reduction (deterministic) ----
    red0[tid] = ce; red1[tid] = l1; red2[tid] = cnt;
    __syncthreads();
    for (int s = THREADS / 2; s > 0; s >>= 1) {
        if (tid < s) {
            red0[tid] += red0[tid + s];
            red1[tid] += red1[tid + s];
            red2[tid] += red2[tid + s];
        }
        __syncthreads();
    }
    if (tid == 0) {
        partial[3 * blockIdx.x + 0] = red0[0];
        partial[3 * blockIdx.x + 1] = red1[0];
        partial[3 * blockIdx.x + 2] = red2[0];
    }
}

__global__ __launch_bounds__(256)
void finalize_kernel(const float* __restrict__ partial, int nblk,
                     float* __restrict__ out) {
    __shared__ float r0[256], r1[256], r2[256];
    int tid = threadIdx.x;
    float a0 = 0.f, a1 = 0.f, a2 = 0.f;
    for (int i = tid; i < nblk; i += 256) {
        a0 += partial[3 * i + 0];
        a1 += partial[3 * i + 1];
        a2 += partial[3 * i + 2];
    }
    r0[tid] = a0; r1[tid] = a1; r2[tid] = a2;
    __syncthreads();
    for (int s = 128; s > 0; s >>= 1) {
        if (tid < s) {
            r0[tid] += r0[tid + s];
            r1[tid] += r1[tid + s];
            r2[tid] += r2[tid + s];
        }
        __syncthreads();
    }
    if (tid == 0) {
        out[0] = r0[0] / (float)((size_t)Bn * Hn * Wn);       // loss_ce (mean BCE)
        float c = r2[0];
        out[1] = (c > 0.f) ? (r1[0] / fmaxf(c, 1.f)) : 0.f;   // masked-L1 loss
    }
}

extern "C" void kernel_launch(void* const* d_in, const int* in_sizes, int n_in,
                              void* d_out, int out_size, void* d_ws, size_t ws_size,
                              hipStream_t stream) {
    const float* obj   = (const float*)d_in[0];   // [B,1,H,W] f32
    const float* lbrts = (const float*)d_in[1];   // [B,4,H,W] f32
    const float* boxes = (const float*)d_in[2];   // [B,N,4]  f32
    float* out     = (float*)d_out;               // [2] f32
    float* partial = (float*)d_ws;                // nblk*3 floats scratch

    const int nblk = Bn * (Hn / ROWS);            // 1024 blocks
    raster_loss_partial<<<nblk, THREADS, 0, stream>>>(obj, lbrts, boxes, partial);
    finalize_kernel<<<1, 256, 0, stream>>>(partial, nblk, out);
}